// PointEncoder_66434554134772
// MI455X (gfx1250) — compile-verified
//
#include <hip/hip_runtime.h>
#include <math.h>

// ---------------------------------------------------------------------------
// PointEncoder for MI455X (gfx1250), fp32 WMMA path (V_WMMA_F32_16X16X4_F32).
// Weights are repacked once per launch into row-pair-interleaved layout so
// every WMMA B-fragment is a single aligned global_load_b64.
// ---------------------------------------------------------------------------

typedef float v2f __attribute__((ext_vector_type(2)));
typedef float v8f __attribute__((ext_vector_type(8)));

#define TWO_PI_F 6.28318530717958647692f

constexpr long long BNP     = 8LL * 65536;           // 524288 points (B*N)
constexpr long long FEAT_SZ = BNP * 64;              // 33554432 floats
constexpr long long SUMS_SZ = 3LL * 8 * 16384 * 64;  // 25165824 floats
constexpr long long CNT_SZ  = 3LL * 8 * 16384;       // 393216 floats
constexpr long long IDX_SZ  = 3LL * BNP;             // ints
// packed weights: w0p 5*8192, w1p 5*4096, wsp 5*8192, pwp 4096
constexpr long long W0P_SZ = 5LL * 8192;
constexpr long long W1P_SZ = 5LL * 4096;
constexpr long long WSP_SZ = 5LL * 8192;
constexpr long long PWP_SZ = 4096;

__device__ __forceinline__ v8f wmma4(v2f a, v2f b, v8f c) {
  // D = A(16x4 f32) * B(4x16 f32) + C(16x16 f32)
  return __builtin_amdgcn_wmma_f32_16x16x4_f32(
      /*neg_a=*/false, a, /*neg_b=*/false, b,
      /*c_mod=*/(short)0, c, /*reuse_a=*/false, /*reuse_b=*/false);
}

// ---------------------------------------------------------------------------
// Weight repack: Wp[(r2*64+col)*2 + j] = W[(2*r2+j)*64 + col]
// (interleave adjacent K rows element-wise -> B fragments become b64 loads)
// ---------------------------------------------------------------------------
__global__ __launch_bounds__(256) void k_pack(
    const float* __restrict__ W, float* __restrict__ Wp, int Khalf, int total)
{
  const int idx = blockIdx.x * 256 + threadIdx.x;   // pair index
  if (idx >= total) return;
  const int per = Khalf * 64;
  const int m   = idx / per;
  const int rem = idx - m * per;
  const int r2  = rem >> 6;
  const int col = rem & 63;
  const long long sbase = (long long)m * (Khalf * 2) * 64;
  const long long dbase = (long long)m * per * 2;
  Wp[dbase + ((long long)(r2 * 64 + col) << 1) + 0] = W[sbase + (2 * r2) * 64 + col];
  Wp[dbase + ((long long)(r2 * 64 + col) << 1) + 1] = W[sbase + (2 * r2 + 1) * 64 + col];
}

// Fused ResnetBlockFC on one 16-point tile held in LDS (x: 16x128).
//   h = relu(relu(x) @ W0 + b0)       (GEMM1, K=128)
//   y = h @ W1 + b1 + x @ Ws          (GEMM2 K=64 + GEMM3 K=128, same acc)
// W0p/W1p/Wsp are the packed layouts.
__device__ __forceinline__ void resnet_tile(
    const float (*xrow)[132], float (*hrow)[68],
    const float* __restrict__ W0p, const float* __restrict__ B0,
    const float* __restrict__ W1p, const float* __restrict__ B1,
    const float* __restrict__ Wsp,
    float* __restrict__ out, long long p0, int lane)
{
  const int row  = lane & 15;
  const int kh   = lane >> 4;     // which half of K this lane holds
  const int col  = lane & 15;

  v8f acc[4];
  #pragma unroll
  for (int n = 0; n < 4; ++n) {
    float bv = B0[n * 16 + col];
    #pragma unroll
    for (int r = 0; r < 8; ++r) acc[n][r] = bv;
  }

  // GEMM1: relu(x) @ W0  (K = 128)
  #pragma unroll 4
  for (int k0 = 0; k0 < 128; k0 += 4) {
    const int ka = k0 + 2 * kh;
    const int r2 = (k0 >> 1) + kh;
    v2f a;
    a.x = fmaxf(xrow[row][ka], 0.0f);
    a.y = fmaxf(xrow[row][ka + 1], 0.0f);
    #pragma unroll
    for (int n = 0; n < 4; ++n) {
      const v2f b = *(const v2f*)(W0p + ((long long)r2 << 7) + ((n * 16 + col) << 1));
      acc[n] = wmma4(a, b, acc[n]);
    }
  }

  // relu(h) -> LDS so we can rebuild row-major A fragments
  #pragma unroll
  for (int n = 0; n < 4; ++n)
    #pragma unroll
    for (int r = 0; r < 8; ++r)
      hrow[r + 8 * kh][n * 16 + col] = fmaxf(acc[n][r], 0.0f);
  __syncthreads();

  // acc = b1; GEMM2: h @ W1 (K=64); GEMM3: x @ Ws (K=128) — accumulate both
  #pragma unroll
  for (int n = 0; n < 4; ++n) {
    float bv = B1[n * 16 + col];
    #pragma unroll
    for (int r = 0; r < 8; ++r) acc[n][r] = bv;
  }
  #pragma unroll 4
  for (int k0 = 0; k0 < 64; k0 += 4) {
    const int ka = k0 + 2 * kh;
    const int r2 = (k0 >> 1) + kh;
    v2f a;
    a.x = hrow[row][ka];
    a.y = hrow[row][ka + 1];
    #pragma unroll
    for (int n = 0; n < 4; ++n) {
      const v2f b = *(const v2f*)(W1p + ((long long)r2 << 7) + ((n * 16 + col) << 1));
      acc[n] = wmma4(a, b, acc[n]);
    }
  }
  #pragma unroll 4
  for (int k0 = 0; k0 < 128; k0 += 4) {
    const int ka = k0 + 2 * kh;
    const int r2 = (k0 >> 1) + kh;
    v2f a;
    a.x = xrow[row][ka];
    a.y = xrow[row][ka + 1];
    #pragma unroll
    for (int n = 0; n < 4; ++n) {
      const v2f b = *(const v2f*)(Wsp + ((long long)r2 << 7) + ((n * 16 + col) << 1));
      acc[n] = wmma4(a, b, acc[n]);
    }
  }

  // D layout: VGPR r holds rows r (lanes 0-15) and r+8 (lanes 16-31)
  #pragma unroll
  for (int n = 0; n < 4; ++n)
    #pragma unroll
    for (int r = 0; r < 8; ++r)
      out[(p0 + r + 8 * kh) * 64 + n * 16 + col] = acc[n][r];
}

// ---------------------------------------------------------------------------
// Kernel 1: RFF embedding + grid indices + per-cell counts + block 0
// ---------------------------------------------------------------------------
__global__ __launch_bounds__(128) void k_rff_block0(
    const float* __restrict__ pts, const float* __restrict__ rff,
    const float* __restrict__ W0p, const float* __restrict__ B0,
    const float* __restrict__ W1p, const float* __restrict__ B1,
    const float* __restrict__ Wsp,
    float* __restrict__ feat, int* __restrict__ idxb, float* __restrict__ cnts)
{
  __shared__ float xs[4][16][132];
  __shared__ float hs[4][16][68];
  const int lane = threadIdx.x & 31;
  const int w    = threadIdx.x >> 5;
  const long long p0 = ((long long)blockIdx.x * 4 + w) * 16;

  __builtin_prefetch(W0p + (threadIdx.x << 6), 0, 1);   // warm WGP$ (32KB)
  __builtin_prefetch(Wsp + (threadIdx.x << 6), 0, 1);

  // Stage: RFF features into LDS (2 cols per lane), indices by lanes 0-15
  const int c = lane * 2;
  for (int m = 0; m < 16; ++m) {
    const long long p = p0 + m;
    const float px = pts[p * 3 + 0], py = pts[p * 3 + 1], pz = pts[p * 3 + 2];
    #pragma unroll
    for (int j = 0; j < 2; ++j) {
      const int cc = c + j;
      const float pr = TWO_PI_F * (px * rff[cc] + py * rff[64 + cc] + pz * rff[128 + cc]);
      xs[w][m][cc]      = __sinf(pr);
      xs[w][m][64 + cc] = __cosf(pr);
    }
  }
  if (lane < 16) {
    const long long p = p0 + lane;
    const int b = (int)(p >> 16);
    const float inv = 1.0f / 1.101f;  // 1 + PAD + 1e-3
    const float fx = (pts[p * 3 + 0] * inv + 1.0f) * 64.0f;
    const float fy = (pts[p * 3 + 1] * inv + 1.0f) * 64.0f;
    const float fz = (pts[p * 3 + 2] * inv + 1.0f) * 64.0f;
    const int ix = (int)fminf(fmaxf(fx, 0.0f), 127.0f);
    const int iy = (int)fminf(fmaxf(fy, 0.0f), 127.0f);
    const int iz = (int)fminf(fmaxf(fz, 0.0f), 127.0f);
    const int i0 = ix + 128 * iy;
    const int i1 = ix + 128 * iz;
    const int i2 = iy + 128 * iz;
    idxb[0 * BNP + p] = i0;
    idxb[1 * BNP + p] = i1;
    idxb[2 * BNP + p] = i2;
    atomicAdd(&cnts[((long long)(0 * 8 + b) << 14) + i0], 1.0f);
    atomicAdd(&cnts[((long long)(1 * 8 + b) << 14) + i1], 1.0f);
    atomicAdd(&cnts[((long long)(2 * 8 + b) << 14) + i2], 1.0f);
  }
  __syncthreads();

  resnet_tile(xs[w], hs[w], W0p, B0, W1p, B1, Wsp, feat, p0, lane);
}

// ---------------------------------------------------------------------------
// Kernel 2: scatter-add features into plane sums.
// One WAVE per (plane, point); each lane adds 2 consecutive channels ->
// atomics from a wave hit 64 consecutive floats of one cell row (L2-friendly).
// ---------------------------------------------------------------------------
__global__ __launch_bounds__(256) void k_scatter_add(
    const float* __restrict__ feat, const int* __restrict__ idxb,
    float* __restrict__ sums)
{
  const long long gtid = (long long)blockIdx.x * 256 + threadIdx.x;
  const long long wid  = gtid >> 5;          // (plane, point)
  if (wid >= 3 * BNP) return;
  const int lane = threadIdx.x & 31;
  const int pl   = (int)(wid >> 19);         // BNP == 2^19
  const long long p = wid & (BNP - 1);
  const int b    = (int)(p >> 16);
  const int cell = idxb[(long long)pl * BNP + p];
  const float2 v = *(const float2*)(feat + p * 64 + lane * 2);
  float* dst = sums + ((((long long)(pl * 8 + b) << 14) + cell) << 6) + lane * 2;
  atomicAdd(dst + 0, v.x);
  atomicAdd(dst + 1, v.y);
}

// ---------------------------------------------------------------------------
// Kernel 3: gather pooled means + resnet block i (in-place feat update)
// ---------------------------------------------------------------------------
__global__ __launch_bounds__(128) void k_resnet_pooled(
    float* __restrict__ feat,
    const float* __restrict__ sums, const float* __restrict__ cnts,
    const int* __restrict__ idxb,
    const float* __restrict__ W0p, const float* __restrict__ B0,
    const float* __restrict__ W1p, const float* __restrict__ B1,
    const float* __restrict__ Wsp)
{
  __shared__ float xs[4][16][132];
  __shared__ float hs[4][16][68];
  const int lane = threadIdx.x & 31;
  const int w    = threadIdx.x >> 5;
  const long long p0 = ((long long)blockIdx.x * 4 + w) * 16;

  __builtin_prefetch(W0p + (threadIdx.x << 6), 0, 1);
  __builtin_prefetch(Wsp + (threadIdx.x << 6), 0, 1);

  const int c = lane * 2;
  for (int m = 0; m < 16; ++m) {
    const long long p = p0 + m;
    const int b = (int)(p >> 16);
    const float2 fv = *(const float2*)(feat + p * 64 + c);
    float pl0 = 0.0f, pl1 = 0.0f;
    #pragma unroll
    for (int pl = 0; pl < 3; ++pl) {
      const int cell = idxb[(long long)pl * BNP + p];
      const long long cb = ((long long)(pl * 8 + b) << 14) + cell;
      const float invc = 1.0f / fmaxf(cnts[cb], 1.0f);
      const float2 sv = *(const float2*)(sums + (cb << 6) + c);
      pl0 += sv.x * invc;
      pl1 += sv.y * invc;
    }
    xs[w][m][c]      = fv.x;
    xs[w][m][c + 1]  = fv.y;
    xs[w][m][64 + c] = pl0;
    xs[w][m][65 + c] = pl1;
  }
  __syncthreads();

  resnet_tile(xs[w], hs[w], W0p, B0, W1p, B1, Wsp, feat, p0, lane);
}

// ---------------------------------------------------------------------------
// Kernel 4: final projection feat @ proj_w + proj_b  (WMMA, K = 64)
// ---------------------------------------------------------------------------
__global__ __launch_bounds__(128) void k_proj(
    const float* __restrict__ feat,
    const float* __restrict__ PWp, const float* __restrict__ PB,
    float* __restrict__ out_feat)
{
  const int lane = threadIdx.x & 31;
  const int w    = threadIdx.x >> 5;
  const long long p0 = ((long long)blockIdx.x * 4 + w) * 16;
  const int row = lane & 15;
  const int kh  = lane >> 4;
  const int col = lane & 15;

  v8f acc[4];
  #pragma unroll
  for (int n = 0; n < 4; ++n) {
    float bv = PB[n * 16 + col];
    #pragma unroll
    for (int r = 0; r < 8; ++r) acc[n][r] = bv;
  }
  #pragma unroll 4
  for (int k0 = 0; k0 < 64; k0 += 4) {
    const int ka = k0 + 2 * kh;
    const int r2 = (k0 >> 1) + kh;
    const v2f a = *(const v2f*)(feat + (p0 + row) * 64 + ka);
    #pragma unroll
    for (int n = 0; n < 4; ++n) {
      const v2f b = *(const v2f*)(PWp + ((long long)r2 << 7) + ((n * 16 + col) << 1));
      acc[n] = wmma4(a, b, acc[n]);
    }
  }
  #pragma unroll
  for (int n = 0; n < 4; ++n)
    #pragma unroll
    for (int r = 0; r < 8; ++r)
      out_feat[(p0 + r + 8 * kh) * 64 + n * 16 + col] = acc[n][r];
}

// ---------------------------------------------------------------------------
// Kernel 5: divide sums by counts + transpose to (plane*B, C, H, W),
// LDS-tiled 64x64 so both read and write sides are coalesced.
// ---------------------------------------------------------------------------
__global__ __launch_bounds__(256) void k_out_planes(
    const float* __restrict__ sums, const float* __restrict__ cnts,
    float* __restrict__ out)
{
  __shared__ float tile[64][65];
  __shared__ float icnt[64];
  const int pb = blockIdx.x >> 8;          // plane*8 + b  (0..23)
  const int ct = blockIdx.x & 255;         // 64-cell tile index
  const long long cell0 = (long long)ct * 64;
  const float* src = sums + ((((long long)pb << 14) + cell0) << 6);
  const int t = threadIdx.x;
  #pragma unroll
  for (int i = 0; i < 16; ++i) {
    const int idx = t + i * 256;           // idx = cell_in_tile*64 + c
    tile[idx >> 6][idx & 63] = src[idx];
  }
  if (t < 64)
    icnt[t] = 1.0f / fmaxf(cnts[((long long)pb << 14) + cell0 + t], 1.0f);
  __syncthreads();
  float* dst = out + (((long long)pb * 64) << 14) + cell0;
  #pragma unroll
  for (int i = 0; i < 16; ++i) {
    const int idx = t + i * 256;           // idx = c*64 + cell_in_tile
    const int c = idx >> 6, ce = idx & 63;
    dst[((long long)c << 14) + ce] = tile[ce][c] * icnt[ce];
  }
}

// ---------------------------------------------------------------------------
extern "C" void kernel_launch(void* const* d_in, const int* in_sizes, int n_in,
                              void* d_out, int out_size, void* d_ws, size_t ws_size,
                              hipStream_t stream) {
  const float* pts = (const float*)d_in[0];
  const float* rff = (const float*)d_in[1];
  const float* w0a = (const float*)d_in[2];
  const float* b0a = (const float*)d_in[3];
  const float* w1a = (const float*)d_in[4];
  const float* b1a = (const float*)d_in[5];
  const float* wsa = (const float*)d_in[6];
  const float* pw  = (const float*)d_in[7];
  const float* pb  = (const float*)d_in[8];

  float* out        = (float*)d_out;
  float* out_planes = out;                       // 25,165,824 floats
  float* out_feat   = out + SUMS_SZ;             // 33,554,432 floats

  float* feat = (float*)d_ws;
  float* sums = feat + FEAT_SZ;
  float* cnts = sums + SUMS_SZ;
  int*   idxb = (int*)(cnts + CNT_SZ);
  float* w0p  = (float*)(idxb + IDX_SZ);
  float* w1p  = w0p + W0P_SZ;
  float* wsp  = w1p + W1P_SZ;
  float* pwp  = wsp + WSP_SZ;

  const int tile_blocks = (int)(BNP / 64);            // 8192 blocks of 4 waves
  const int sc_blocks   = (int)(3 * BNP * 32 / 256);  // 196608
  const int tp_blocks   = 24 * 256;                   // 6144

  hipMemsetAsync(cnts, 0, (size_t)CNT_SZ * 4, stream);

  // pack weights: pairs = nmat * (K/2) * 64
  k_pack<<<(5 * 64 * 64 + 255) / 256, 256, 0, stream>>>(w0a, w0p, 64, 5 * 64 * 64);
  k_pack<<<(5 * 32 * 64 + 255) / 256, 256, 0, stream>>>(w1a, w1p, 32, 5 * 32 * 64);
  k_pack<<<(5 * 64 * 64 + 255) / 256, 256, 0, stream>>>(wsa, wsp, 64, 5 * 64 * 64);
  k_pack<<<(1 * 32 * 64 + 255) / 256, 256, 0, stream>>>(pw,  pwp, 32, 1 * 32 * 64);

  k_rff_block0<<<tile_blocks, 128, 0, stream>>>(
      pts, rff, w0p, b0a, w1p, b1a, wsp, feat, idxb, cnts);

  for (int i = 1; i < 5; ++i) {
    hipMemsetAsync(sums, 0, (size_t)SUMS_SZ * 4, stream);
    k_scatter_add<<<sc_blocks, 256, 0, stream>>>(feat, idxb, sums);
    k_resnet_pooled<<<tile_blocks, 128, 0, stream>>>(
        feat, sums, cnts, idxb,
        w0p + (long long)i * 8192, b0a + (long long)i * 64,
        w1p + (long long)i * 4096, b1a + (long long)i * 64,
        wsp + (long long)i * 8192);
  }

  k_proj<<<tile_blocks, 128, 0, stream>>>(feat, pwp, pb, out_feat);

  hipMemsetAsync(sums, 0, (size_t)SUMS_SZ * 4, stream);
  k_scatter_add<<<sc_blocks, 256, 0, stream>>>(out_feat, idxb, sums);
  k_out_planes<<<tp_blocks, 256, 0, stream>>>(sums, cnts, out_planes);
}